// GATODEFunc_17763984736722
// MI455X (gfx1250) — compile-verified
//
#include <hip/hip_runtime.h>
#include <math.h>

#define N_NODES   50000
#define N_EDGES   800000
#define HID       64
#define HEADS     4
#define CPH       64              // channels per head
#define FPROJ     (HEADS * CPH)   // 256
#define EDIM      3
#define LAYERS    2
#define NEG_SLOPE 0.2f
#define LN_EPS    1e-5f

static_assert(N_NODES % 16 == 0, "row tiles exact");
static_assert(N_NODES % 4 == 0, "finalize tiles exact");

typedef __attribute__((ext_vector_type(2))) float v2f;
typedef __attribute__((ext_vector_type(8))) float v8f;

// ---------- ordered-float encoding for atomic max ----------
__device__ __forceinline__ unsigned fenc(float x) {
    unsigned u = __float_as_uint(x);
    return (u & 0x80000000u) ? ~u : (u | 0x80000000u);
}
__device__ __forceinline__ float fdec(unsigned u) {
    return (u & 0x80000000u) ? __uint_as_float(u & 0x7FFFFFFFu)
                             : __uint_as_float(~u);
}
#define MENC_NEG_INF 0x007FFFFFu   // == fenc(-inf)

// ---------------------------------------------------------------------------
// Kernel 1: node projection X = Z @ W   ([N,64] x [64,256]) via f32 WMMA.
// Block = 256 threads (8 waves). 16-row A tile staged in LDS; wave w computes
// column tiles w and w+8; K-loop = 16 x V_WMMA_F32_16X16X4_F32.
// N_NODES % 16 == 0 -> branch-free store epilogue (EXEC stays all-ones).
// ---------------------------------------------------------------------------
__global__ __launch_bounds__(256)
void node_proj_wmma(const float* __restrict__ Z, const float* __restrict__ W,
                    float* __restrict__ X) {
    __shared__ float As[16][68];                 // +4 pad: avoid bank conflicts
    const int rowBase = blockIdx.x * 16;
    const int tid = threadIdx.x;
    for (int i = tid; i < 16 * HID; i += 256) {  // stage 16x64 A tile
        int r = i >> 6, k = i & 63;
        As[r][k] = Z[(rowBase + r) * HID + k];
    }
    __syncthreads();

    const int wave = tid >> 5;
    const int lane = tid & 31;
    const int half = lane >> 4;   // 0: K{0,1} lanes, 1: K{2,3} lanes
    const int l15  = lane & 15;
    const int kOff = half * 2;

    #pragma unroll
    for (int ctStep = 0; ctStep < 2; ++ctStep) {
        const int ct  = wave + ctStep * 8;       // column tile 0..15
        const int col = ct * 16 + l15;
        v8f acc = {};
        #pragma unroll
        for (int k0 = 0; k0 < HID; k0 += 4) {
            v2f a, b;
            // A 16x4: lanes 0-15 rows M=l15 hold K=k0,k0+1; lanes 16-31 K=k0+2,k0+3
            a.x = As[l15][k0 + kOff];
            a.y = As[l15][k0 + kOff + 1];
            // B 4x16: K rows striped across VGPRs, N across lanes
            b.x = W[(k0 + kOff)     * FPROJ + col];
            b.y = W[(k0 + kOff + 1) * FPROJ + col];
            acc = __builtin_amdgcn_wmma_f32_16x16x4_f32(
                false, a, false, b, (short)0, acc, false, false);
        }
        #pragma unroll
        for (int r = 0; r < 8; ++r)              // C/D: vgpr r -> M = r (+8)
            X[(rowBase + r + half * 8) * FPROJ + col] = acc[r];
    }
}

// ---------------------------------------------------------------------------
// Kernel 2: per-(node,head) attention scalars  as = <x_h, att_src_h>, ad = ...
// ---------------------------------------------------------------------------
__global__ __launch_bounds__(256)
void node_alpha(const float* __restrict__ X, const float* __restrict__ a_s,
                const float* __restrict__ a_d, float* __restrict__ as_out,
                float* __restrict__ ad_out) {
    int i = blockIdx.x * blockDim.x + threadIdx.x;
    if (i >= N_NODES * HEADS) return;
    int node = i / HEADS, h = i % HEADS;
    const float* xp = X + node * FPROJ + h * CPH;
    const float* sp = a_s + h * CPH;
    const float* dp = a_d + h * CPH;
    float accs = 0.0f, accd = 0.0f;
    #pragma unroll 8
    for (int c = 0; c < CPH; ++c) { float x = xp[c]; accs += x * sp[c]; accd += x * dp[c]; }
    as_out[i] = accs;
    ad_out[i] = accd;
}

// ---------------------------------------------------------------------------
// Kernel 3: reduced edge matrix P[d][h] = sum_c We[d, h*C+c] * att_edge[h,c]
// (edge features only touch the logit, so [E,3]x[3,256] collapses to 3x4)
// ---------------------------------------------------------------------------
__global__ void edge_reduce_P(const float* __restrict__ We,
                              const float* __restrict__ ae,
                              float* __restrict__ P) {
    int i = threadIdx.x;
    if (i >= EDIM * HEADS) return;
    int d = i / HEADS, h = i % HEADS;
    float s = 0.0f;
    for (int c = 0; c < CPH; ++c)
        s += We[d * FPROJ + h * CPH + c] * ae[h * CPH + c];
    P[i] = s;
}

__global__ void init_menc(unsigned* __restrict__ m_enc) {
    int i = blockIdx.x * blockDim.x + threadIdx.x;
    if (i < N_NODES * HEADS) m_enc[i] = MENC_NEG_INF;
}

// ---------------------------------------------------------------------------
// Kernel 4: edge logits + leaky relu + segment max (ordered-uint atomicMax)
// ---------------------------------------------------------------------------
__global__ __launch_bounds__(256)
void edge_logits(const int* __restrict__ src, const int* __restrict__ dst,
                 const float* __restrict__ ea, const float* __restrict__ P,
                 const float* __restrict__ as_, const float* __restrict__ ad_,
                 float* __restrict__ alphaE, unsigned* __restrict__ m_enc) {
    int i = blockIdx.x * blockDim.x + threadIdx.x;
    if (i >= N_EDGES * HEADS) return;
    int e = i >> 2, h = i & 3;
    int s = src[e], d = dst[e];
    float v = as_[s * HEADS + h] + ad_[d * HEADS + h];
    const float* eaE = ea + e * EDIM;
    #pragma unroll
    for (int k = 0; k < EDIM; ++k) v += eaE[k] * P[k * HEADS + h];
    v = (v > 0.0f) ? v : v * NEG_SLOPE;
    alphaE[i] = v;
    atomicMax(&m_enc[d * HEADS + h], fenc(v));
}

__global__ void m_finalize(const unsigned* __restrict__ m_enc,
                           float* __restrict__ m) {
    int i = blockIdx.x * blockDim.x + threadIdx.x;
    if (i >= N_NODES * HEADS) return;
    float v = fdec(m_enc[i]);
    m[i] = isfinite(v) ? v : 0.0f;   // empty segments -> 0 (matches reference)
}

// ---------------------------------------------------------------------------
// Kernel 5: ex = exp(alpha - m[dst]) (in-place), denom[dst] += ex
// ---------------------------------------------------------------------------
__global__ __launch_bounds__(256)
void edge_exp(const int* __restrict__ dst, float* __restrict__ alphaE,
              const float* __restrict__ m, float* __restrict__ denom) {
    int i = blockIdx.x * blockDim.x + threadIdx.x;
    if (i >= N_EDGES * HEADS) return;
    int e = i >> 2, h = i & 3;
    int d = dst[e];
    float ex = __expf(alphaE[i] - m[d * HEADS + h]);
    alphaE[i] = ex;
    atomicAdd(&denom[d * HEADS + h], ex);
}

// ---------------------------------------------------------------------------
// Kernel 6: head-fused weighted scatter, float4 per thread:
//   acc[dst, c..c+3] += (1/H) * sum_h coef[e,h] * X[src, h*C + c..c+3]
// 16 threads per edge; global_load_b128 gathers; 4 atomics per thread.
// ---------------------------------------------------------------------------
__global__ __launch_bounds__(256)
void edge_scatter(const int* __restrict__ src, const int* __restrict__ dst,
                  const float* __restrict__ ex, const float* __restrict__ denom,
                  const float* __restrict__ X, float* __restrict__ acc) {
    unsigned i = blockIdx.x * 256u + threadIdx.x;   // E * 16 threads
    if (i >= (unsigned)N_EDGES * (CPH / 4)) return;
    int e  = i >> 4;
    int c4 = (i & 15) * 4;
    int s = src[e], d = dst[e];
    float4 sum = {0.0f, 0.0f, 0.0f, 0.0f};
    #pragma unroll
    for (int h = 0; h < HEADS; ++h) {
        float coef = ex[e * HEADS + h] / (denom[d * HEADS + h] + 1e-16f);
        float4 x = *(const float4*)&X[s * FPROJ + h * CPH + c4];
        sum.x += coef * x.x; sum.y += coef * x.y;
        sum.z += coef * x.z; sum.w += coef * x.w;
    }
    float* out = &acc[d * CPH + c4];
    atomicAdd(out + 0, sum.x * (1.0f / HEADS));
    atomicAdd(out + 1, sum.y * (1.0f / HEADS));
    atomicAdd(out + 2, sum.z * (1.0f / HEADS));
    atomicAdd(out + 3, sum.w * (1.0f / HEADS));
}

// ---------------------------------------------------------------------------
// Kernel 7: +bias, LayerNorm(64), SiLU.  4 nodes per 256-thread block.
// ---------------------------------------------------------------------------
__global__ __launch_bounds__(256)
void node_finalize(const float* __restrict__ acc, const float* __restrict__ bias,
                   const float* __restrict__ gamma, const float* __restrict__ beta,
                   float* __restrict__ zout) {
    __shared__ float red[256];
    const int tid = threadIdx.x;
    const int g = tid >> 6;          // node slot in block
    const int c = tid & 63;          // channel
    const int node = blockIdx.x * 4 + g;   // N_NODES % 4 == 0: always valid
    float y = acc[node * CPH + c] + bias[c];

    red[tid] = y; __syncthreads();
    #pragma unroll
    for (int off = 32; off >= 1; off >>= 1) {
        if (c < off) red[tid] += red[tid + off];
        __syncthreads();
    }
    float mu = red[g * 64] * (1.0f / 64.0f);
    __syncthreads();

    float dv = y - mu;
    red[tid] = dv * dv; __syncthreads();
    #pragma unroll
    for (int off = 32; off >= 1; off >>= 1) {
        if (c < off) red[tid] += red[tid + off];
        __syncthreads();
    }
    float var = red[g * 64] * (1.0f / 64.0f);

    float zl = dv * rsqrtf(var + LN_EPS) * gamma[c] + beta[c];
    zout[node * CPH + c] = zl / (1.0f + __expf(-zl));   // SiLU
}

// ---------------------------------------------------------------------------
extern "C" void kernel_launch(void* const* d_in, const int* in_sizes, int n_in,
                              void* d_out, int out_size, void* d_ws, size_t ws_size,
                              hipStream_t stream) {
    (void)in_sizes; (void)n_in; (void)out_size; (void)ws_size;
    const float* h          = (const float*)d_in[1];
    const int*   edge_index = (const int*)d_in[2];
    const float* edge_attr  = (const float*)d_in[3];
    const float* W          = (const float*)d_in[4];   // [L,64,256]
    const float* We         = (const float*)d_in[5];   // [L,3,256]
    const float* att_src    = (const float*)d_in[6];   // [L,4,64]
    const float* att_dst    = (const float*)d_in[7];
    const float* att_edge   = (const float*)d_in[8];
    const float* bias       = (const float*)d_in[9];   // [L,64]
    const float* gamma      = (const float*)d_in[10];
    const float* beta       = (const float*)d_in[11];

    const int* src = edge_index;
    const int* dst = edge_index + N_EDGES;

    // ---- workspace carve (all 256B aligned) ----
    char* p = (char*)d_ws;
    auto carve = [&](size_t bytes) { char* r = p; p += (bytes + 255) & ~size_t(255); return r; };
    float*    X      = (float*)   carve(sizeof(float) * N_NODES * FPROJ);   // 51.2 MB
    float*    as_    = (float*)   carve(sizeof(float) * N_NODES * HEADS);
    float*    ad_    = (float*)   carve(sizeof(float) * N_NODES * HEADS);
    float*    P      = (float*)   carve(sizeof(float) * 16);
    float*    alphaE = (float*)   carve(sizeof(float) * (size_t)N_EDGES * HEADS); // 12.8 MB
    unsigned* m_enc  = (unsigned*)carve(sizeof(unsigned) * N_NODES * HEADS);
    float*    mbuf   = (float*)   carve(sizeof(float) * N_NODES * HEADS);
    float*    denom  = (float*)   carve(sizeof(float) * N_NODES * HEADS);
    float*    accb   = (float*)   carve(sizeof(float) * N_NODES * CPH);     // 12.8 MB
    float*    zbuf   = (float*)   carve(sizeof(float) * N_NODES * CPH);     // 12.8 MB

    const int NH = N_NODES * HEADS;
    const int EH = N_EDGES * HEADS;
    const unsigned EC4 = (unsigned)N_EDGES * (CPH / 4);

    const float* zin = h;
    for (int l = 0; l < LAYERS; ++l) {
        float* zout = (l == LAYERS - 1) ? (float*)d_out : zbuf;

        hipMemsetAsync(accb,  0, sizeof(float) * N_NODES * CPH, stream);
        hipMemsetAsync(denom, 0, sizeof(float) * NH, stream);
        init_menc<<<(NH + 255) / 256, 256, 0, stream>>>(m_enc);

        node_proj_wmma<<<N_NODES / 16, 256, 0, stream>>>(
            zin, W + (size_t)l * HID * FPROJ, X);

        node_alpha<<<(NH + 255) / 256, 256, 0, stream>>>(
            X, att_src + l * HEADS * CPH, att_dst + l * HEADS * CPH, as_, ad_);

        edge_reduce_P<<<1, 16, 0, stream>>>(
            We + (size_t)l * EDIM * FPROJ, att_edge + l * HEADS * CPH, P);

        edge_logits<<<(EH + 255) / 256, 256, 0, stream>>>(
            src, dst, edge_attr, P, as_, ad_, alphaE, m_enc);

        m_finalize<<<(NH + 255) / 256, 256, 0, stream>>>(m_enc, mbuf);

        edge_exp<<<(EH + 255) / 256, 256, 0, stream>>>(dst, alphaE, mbuf, denom);

        edge_scatter<<<(EC4 + 255u) / 256u, 256, 0, stream>>>(
            src, dst, alphaE, denom, X, accb);

        node_finalize<<<N_NODES / 4, 256, 0, stream>>>(
            accb, bias + l * HID, gamma + l * HID, beta + l * HID, zout);

        zin = zout;
    }
}